// NetGIN_20469814132907
// MI455X (gfx1250) — compile-verified
//
#include <hip/hip_runtime.h>
#include <math.h>

// ---------------------------------------------------------------------------
// GIN network for CDNA5 (gfx1250, wave32).
//   - edge scatter-add: one thread per edge, global f32 atomics (HBM bound)
//   - node MLPs: one wave per 16-node tile using V_WMMA_F32_16X16X4_F32,
//     with all operands staged through zero-padded LDS tiles so the WMMA
//     chain runs with EXEC all-1s and no per-fragment predication
//   - graph mean-pool fused into MLP epilogue via f32 atomics
// ---------------------------------------------------------------------------

#define N_NODES_C    200000
#define N_EDGES_C    5000000
#define NUM_GRAPHS_C 1000

typedef __attribute__((ext_vector_type(2))) float v2f;
typedef __attribute__((ext_vector_type(8))) float v8f;

// -------------------------------- utility ----------------------------------

__global__ void zero_f32_kernel(float* __restrict__ p, long n) {
    long i = (long)blockIdx.x * blockDim.x + threadIdx.x;
    long stride = (long)gridDim.x * blockDim.x;
    for (; i < n; i += stride) p[i] = 0.0f;
}

__global__ void count_kernel(const int* __restrict__ batch,
                             float* __restrict__ counts, int n_nodes) {
    int i = blockIdx.x * blockDim.x + threadIdx.x;
    if (i < n_nodes) atomicAdd(&counts[batch[i]], 1.0f);
}

// ------------------------------ scatter-add --------------------------------

// dim-16 gather (layer 1 input x): float4 loads, 16 f32 atomics
__global__ void scatter_add16_kernel(const float* __restrict__ h,
                                     const int* __restrict__ src,
                                     const int* __restrict__ dst,
                                     float* __restrict__ agg, int n_edges) {
    int e = blockIdx.x * blockDim.x + threadIdx.x;
    if (e >= n_edges) return;
    int s = src[e];
    int d = dst[e];
    const float4* hs = (const float4*)(h + (long)s * 16);
    float4 a0 = hs[0], a1 = hs[1], a2 = hs[2], a3 = hs[3];
    float* ad = agg + (long)d * 16;
    atomicAdd(ad + 0,  a0.x); atomicAdd(ad + 1,  a0.y);
    atomicAdd(ad + 2,  a0.z); atomicAdd(ad + 3,  a0.w);
    atomicAdd(ad + 4,  a1.x); atomicAdd(ad + 5,  a1.y);
    atomicAdd(ad + 6,  a1.z); atomicAdd(ad + 7,  a1.w);
    atomicAdd(ad + 8,  a2.x); atomicAdd(ad + 9,  a2.y);
    atomicAdd(ad + 10, a2.z); atomicAdd(ad + 11, a2.w);
    atomicAdd(ad + 12, a3.x); atomicAdd(ad + 13, a3.y);
    atomicAdd(ad + 14, a3.z); atomicAdd(ad + 15, a3.w);
}

// dim-10 gather (layers 2..5): rows are 8B aligned (40B stride) -> float2 x5
__global__ void scatter_add10_kernel(const float* __restrict__ h,
                                     const int* __restrict__ src,
                                     const int* __restrict__ dst,
                                     float* __restrict__ agg, int n_edges) {
    int e = blockIdx.x * blockDim.x + threadIdx.x;
    if (e >= n_edges) return;
    int s = src[e];
    int d = dst[e];
    const float2* hs = (const float2*)(h + (long)s * 10);
    float2 v0 = hs[0], v1 = hs[1], v2 = hs[2], v3 = hs[3], v4 = hs[4];
    float* ad = agg + (long)d * 10;
    atomicAdd(ad + 0, v0.x); atomicAdd(ad + 1, v0.y);
    atomicAdd(ad + 2, v1.x); atomicAdd(ad + 3, v1.y);
    atomicAdd(ad + 4, v2.x); atomicAdd(ad + 5, v2.y);
    atomicAdd(ad + 6, v3.x); atomicAdd(ad + 7, v3.y);
    atomicAdd(ad + 8, v4.x); atomicAdd(ad + 9, v4.y);
}

// ----------------------------- WMMA node MLP -------------------------------
//
// One wave handles a 16-node tile.  z = h + agg  (16 x KIN, KIN in {16,10}),
//   mid = relu(z @ W1 + b1)          (16 x 10, K = KIN padded to mult of 4)
//   out = relu(mid @ W2 + b2)        (16 x 10, K = 10 padded to 12)
// out is accumulated into pool[batch[node]] for the mean-pool.
//
// Because the LDS-staged weight tiles have columns >= 10 zeroed and the
// staged biases are zero there, D columns >= 10 are exactly 0 after
// fmaxf(v + bias, 0) — no column predicate is needed in the ReLU.
//
// f32 WMMA operand layouts (ISA 7.12.2, 32-bit, wave32):
//   A 16x4:  lane L(0..15): v0=A[L][0], v1=A[L][1];  lane L+16: v0=A[L][2], v1=A[L][3]
//   B 4x16:  mirrored K-split across lane halves
//   C/D 16x16: vgpr r, lanes 0-15 -> (M=r, N=lane), lanes 16-31 -> (M=r+8, N=lane-16)

template <int KIN>
__global__ void gin_mlp_kernel(const float* __restrict__ h,    // [N, KIN]
                               const float* __restrict__ agg,  // [N, KIN]
                               const float* __restrict__ W1,   // [KIN, 10]
                               const float* __restrict__ B1,   // [10]
                               const float* __restrict__ W2,   // [10, 10]
                               const float* __restrict__ B2,   // [10]
                               float* __restrict__ out,        // [N, 10]
                               float* __restrict__ pool,       // [G, 10]
                               const int* __restrict__ batch,  // [N]
                               int n_nodes) {
    __shared__ float w1s[16][16];       // zero-padded W1
    __shared__ float w2s[12][16];       // zero-padded W2
    __shared__ float b1s[16];
    __shared__ float b2s[16];
    __shared__ float zt[8][16][17];     // per-wave 16x16 tile (padded stride)

    const int tid  = threadIdx.x;
    const int lane = tid & 31;
    const int wave = tid >> 5;
    const int row  = lane & 15;    // A layout: node row in tile
    const int kh   = lane >> 4;    // K half-select
    const int n    = row;          // B/C/D layout: output column
    const int base = (blockIdx.x * 8 + wave) * 16;

    // ---- stage zero-padded weights + biases into LDS -----------------
    {
        const int r  = tid >> 4;
        const int cn = tid & 15;
        w1s[r][cn] = (r < KIN && cn < 10) ? W1[r * 10 + cn] : 0.0f;
        if (tid < 192)
            w2s[r][cn] = (r < 10 && cn < 10) ? W2[r * 10 + cn] : 0.0f;
        if (tid < 16) {
            b1s[tid] = (tid < 10) ? B1[tid] : 0.0f;
            b2s[tid] = (tid < 10) ? B2[tid] : 0.0f;
        }
    }

    // ---- stage z = h + agg into per-wave LDS tile (zero-padded) ------
    {
        constexpr int CPL = KIN / 2;            // cols per lane-half (8 or 5)
        constexpr int PAD = (16 - KIN) / 2;     // pad cols per lane-half (0 or 3)
        const int r = lane >> 1;
        const int node = base + r;
        const int colbase = (lane & 1) * CPL;
        if (node < n_nodes) {
            const float* hp = h + (long)node * KIN + colbase;
            const float* ap = agg + (long)node * KIN + colbase;
#pragma unroll
            for (int i = 0; i < CPL; ++i)
                zt[wave][r][colbase + i] = hp[i] + ap[i];
        } else {
#pragma unroll
            for (int i = 0; i < CPL; ++i)
                zt[wave][r][colbase + i] = 0.0f;
        }
#pragma unroll
        for (int i = 0; i < PAD; ++i)
            zt[wave][r][KIN + (lane & 1) * PAD + i] = 0.0f;
    }
    __syncthreads();

    // ---- first linear: z @ W1 (K = KIN padded to mult of 4) ----------
    constexpr int KFRAG = (KIN + 3) / 4;
    v8f c = {};
#pragma unroll
    for (int j = 0; j < KFRAG; ++j) {
        const int k0 = 4 * j + 2 * kh;
        v2f a, b;
        a.x = zt[wave][row][k0];
        a.y = zt[wave][row][k0 + 1];
        b.x = w1s[k0][n];
        b.y = w1s[k0 + 1][n];
        c = __builtin_amdgcn_wmma_f32_16x16x4_f32(false, a, false, b,
                                                  (short)0, c, false, false);
    }

    // bias + ReLU; write mid tile back into zt (padding cols are 0 already)
    {
        const float bias1 = b1s[n];
#pragma unroll
        for (int r = 0; r < 8; ++r)
            zt[wave][r + 8 * kh][n] = fmaxf(c[r] + bias1, 0.0f);
    }
    __syncthreads();

    // ---- second linear: mid @ W2 (K = 10 padded to 12) ---------------
    v8f c2 = {};
#pragma unroll
    for (int j = 0; j < 3; ++j) {
        const int k0 = 4 * j + 2 * kh;
        v2f a, b;
        a.x = zt[wave][row][k0];
        a.y = zt[wave][row][k0 + 1];
        b.x = w2s[k0][n];
        b.y = w2s[k0 + 1][n];
        c2 = __builtin_amdgcn_wmma_f32_16x16x4_f32(false, a, false, b,
                                                   (short)0, c2, false, false);
    }
    __syncthreads();   // all reads of mid tile done before overwrite

    // bias + outer ReLU -> LDS for a coalesced, uniform epilogue
    {
        const float bias2 = b2s[n];
#pragma unroll
        for (int r = 0; r < 8; ++r)
            zt[wave][r + 8 * kh][n] = fmaxf(c2[r] + bias2, 0.0f);
    }
    __syncthreads();

    // ---- epilogue: store x_next + fused mean-pool atomics ------------
    {
        const int r = lane >> 1;
        const int node = base + r;
        const int colbase = (lane & 1) * 5;   // 5 cols per lane-half
        if (node < n_nodes) {
            const int bg = batch[node];
            float* op = out + (long)node * 10 + colbase;
            float* pp = pool + (long)bg * 10 + colbase;
#pragma unroll
            for (int i = 0; i < 5; ++i) {
                float v = zt[wave][r][colbase + i];
                op[i] = v;
                atomicAdd(pp + i, v);
            }
        }
    }
}

// ------------------------------ final head ---------------------------------

__global__ void final_kernel(const float* __restrict__ pool,   // [5, G, 10]
                             const float* __restrict__ counts, // [G]
                             const float* __restrict__ lw,     // [5, 10]
                             float* __restrict__ out,          // [G]
                             int n_graphs) {
    int g = blockIdx.x * blockDim.x + threadIdx.x;
    if (g >= n_graphs) return;
    float cnt = counts[g];
    float inv = 1.0f / (cnt > 1.0f ? cnt : 1.0f);
    float acc = 0.0f;
#pragma unroll
    for (int l = 0; l < 5; ++l) {
#pragma unroll
        for (int d = 0; d < 10; ++d) {
            acc += pool[((long)l * n_graphs + g) * 10 + d] * lw[l * 10 + d];
        }
    }
    acc *= 0.0f + inv;   // scale once: sum_l l_i(mean) == inv * sum_l l_i(sum)
    out[g] = 1.0f / (1.0f + expf(-acc));
}

// ------------------------------- launcher ----------------------------------

extern "C" void kernel_launch(void* const* d_in, const int* in_sizes, int n_in,
                              void* d_out, int out_size, void* d_ws, size_t ws_size,
                              hipStream_t stream) {
    const float* x     = (const float*)d_in[0];            // [N, 16]
    const int*   eidx  = (const int*)d_in[1];              // [2, E]
    const int*   batch = (const int*)d_in[2];              // [N]
    const float* w1_1  = (const float*)d_in[3];            // [16, 10]
    const float* b1_1  = (const float*)d_in[4];            // [10]
    const float* w2_1  = (const float*)d_in[5];            // [10, 10]
    const float* b2_1  = (const float*)d_in[6];            // [10]
    const float* ws1   = (const float*)d_in[7];            // [4, 10, 10]
    const float* bs1   = (const float*)d_in[8];            // [4, 10]
    const float* ws2   = (const float*)d_in[9];            // [4, 10, 10]
    const float* bs2   = (const float*)d_in[10];           // [4, 10]
    const float* lw    = (const float*)d_in[11];           // [5, 10]
    float* out = (float*)d_out;                            // [G]

    const int N = N_NODES_C;
    const int E = N_EDGES_C;
    const int G = NUM_GRAPHS_C;

    const int* src = eidx;
    const int* dst = eidx + E;

    // workspace layout (floats)
    float* agg    = (float*)d_ws;               // N*16
    float* hA     = agg + (long)N * 16;         // N*10
    float* hB     = hA + (long)N * 10;          // N*10
    float* pool   = hB + (long)N * 10;          // 5*G*10
    float* counts = pool + (long)5 * G * 10;    // G

    const int ZB = 256;
    const int zgrid = 2048;

    // zero pool + counts (contiguous)
    zero_f32_kernel<<<zgrid, ZB, 0, stream>>>(pool, (long)5 * G * 10 + G);
    // per-graph node counts
    count_kernel<<<(N + 255) / 256, 256, 0, stream>>>(batch, counts, N);

    const int egrid = (E + 255) / 256;
    const int mlp_blocks = (N + 127) / 128;     // 8 waves/block * 16 nodes/wave

    // ---- layer 1 (dim 16 -> 10) ----
    zero_f32_kernel<<<zgrid, ZB, 0, stream>>>(agg, (long)N * 16);
    scatter_add16_kernel<<<egrid, 256, 0, stream>>>(x, src, dst, agg, E);
    gin_mlp_kernel<16><<<mlp_blocks, 256, 0, stream>>>(
        x, agg, w1_1, b1_1, w2_1, b2_1, hA, pool + 0, batch, N);

    // ---- layers 2..5 (dim 10 -> 10) ----
    float* hin = hA;
    float* hout = hB;
    for (int l = 0; l < 4; ++l) {
        zero_f32_kernel<<<zgrid, ZB, 0, stream>>>(agg, (long)N * 10);
        scatter_add10_kernel<<<egrid, 256, 0, stream>>>(hin, src, dst, agg, E);
        gin_mlp_kernel<10><<<mlp_blocks, 256, 0, stream>>>(
            hin, agg,
            ws1 + (long)l * 100, bs1 + (long)l * 10,
            ws2 + (long)l * 100, bs2 + (long)l * 10,
            hout, pool + (long)(l + 1) * G * 10, batch, N);
        float* t = hin; hin = hout; hout = t;
    }

    // ---- pooled readout ----
    final_kernel<<<(G + 255) / 256, 256, 0, stream>>>(pool, counts, lw, out, G);
}